// Ising_3470333575731
// MI455X (gfx1250) — compile-verified
//
#include <hip/hip_runtime.h>

typedef __attribute__((ext_vector_type(2))) float v2f;
typedef __attribute__((ext_vector_type(8))) float v8f;

#define NSITES   4096
#define NGRID    64
#define NBATCH   512
#define ROWTILE  16
#define KCHUNKS  8
#define CHUNK    (NSITES / KCHUNKS)          /* 512 sites per block */
#define THREADS  256
#define WAVES    (THREADS / 32)
#define SITES_PER_LANE (CHUNK / (WAVES * 2)) /* 32 */

__global__ void ising_zero(float* __restrict__ out) {
    out[threadIdx.x] = 0.0f;
}

// Gather the two nonzero diagonals of (binary * mask) into compact vectors.
__global__ void ising_precomp(const float* __restrict__ binary,
                              const float* __restrict__ mask,
                              float* __restrict__ wr,
                              float* __restrict__ wd) {
    int i = blockIdx.x * blockDim.x + threadIdx.x;
    if (i >= NSITES) return;
    float r = 0.0f, d = 0.0f;
    if (i < NSITES - 1 && ((i & (NGRID - 1)) != (NGRID - 1))) {
        size_t idx = (size_t)i * NSITES + i + 1;
        r = binary[idx] * mask[idx];
    }
    if (i < NSITES - NGRID) {
        size_t idx = (size_t)i * NSITES + i + NGRID;
        d = binary[idx] * mask[idx];
    }
    wr[i] = r;
    wd[i] = d;
}

__device__ __forceinline__ float coef_r(const float* __restrict__ b,
                                        const float* __restrict__ m, int i) {
    if (i < NSITES - 1 && ((i & (NGRID - 1)) != (NGRID - 1))) {
        size_t idx = (size_t)i * NSITES + i + 1;
        return b[idx] * m[idx];
    }
    return 0.0f;
}

__device__ __forceinline__ float coef_d(const float* __restrict__ b,
                                        const float* __restrict__ m, int i) {
    if (i < NSITES - NGRID) {
        size_t idx = (size_t)i * NSITES + i + NGRID;
        return b[idx] * m[idx];
    }
    return 0.0f;
}

__global__ __launch_bounds__(THREADS) void ising_energy(
    const float* __restrict__ x, const float* __restrict__ unary,
    const float* __restrict__ binary, const float* __restrict__ mask,
    const float* __restrict__ wr, const float* __restrict__ wd,
    int use_pre, float* __restrict__ out)
{
    const int lane    = threadIdx.x & 31;
    const int wave    = threadIdx.x >> 5;
    const int rowTile = blockIdx.x / KCHUNKS;
    const int chunk   = blockIdx.x % KCHUNKS;
    const int row     = rowTile * ROWTILE + (lane & 15);
    const int grp     = wave * 2 + (lane >> 4);             // 0..15
    const int i0      = chunk * CHUNK + grp * SITES_PER_LANE;

    const float* __restrict__ xrow = x + (size_t)row * NSITES;

    float acc = 0.0f;
#pragma unroll
    for (int ii = 0; ii < SITES_PER_LANE; ii += 4) {
        const int i = i0 + ii;
        const float4 xv = *(const float4*)(xrow + i);
        const float  xn = (i + 4 < NSITES) ? xrow[i + 4] : 0.0f;
        float4 xd = make_float4(0.0f, 0.0f, 0.0f, 0.0f);
        if (i + NGRID + 4 <= NSITES)
            xd = *(const float4*)(xrow + i + NGRID);
        const float4 u4 = *(const float4*)(unary + i);
        float4 r4, d4;
        if (use_pre) {
            r4 = *(const float4*)(wr + i);
            d4 = *(const float4*)(wd + i);
        } else {
            r4.x = coef_r(binary, mask, i);     d4.x = coef_d(binary, mask, i);
            r4.y = coef_r(binary, mask, i + 1); d4.y = coef_d(binary, mask, i + 1);
            r4.z = coef_r(binary, mask, i + 2); d4.z = coef_d(binary, mask, i + 2);
            r4.w = coef_r(binary, mask, i + 3); d4.w = coef_d(binary, mask, i + 3);
        }
        acc = fmaf(xv.x, fmaf(r4.x, xv.y, fmaf(d4.x, xd.x, u4.x)), acc);
        acc = fmaf(xv.y, fmaf(r4.y, xv.z, fmaf(d4.y, xd.y, u4.y)), acc);
        acc = fmaf(xv.z, fmaf(r4.z, xv.w, fmaf(d4.z, xd.z, u4.z)), acc);
        acc = fmaf(xv.w, fmaf(r4.w, xn,   fmaf(d4.w, xd.w, u4.w)), acc);
    }

    // Cross-wave reduction via LDS: combine the 8 waves at matching lane slots.
    __shared__ float red[THREADS];
    red[threadIdx.x] = acc;
    __syncthreads();
    float p = 0.0f;
#pragma unroll
    for (int w = 0; w < WAVES; ++w) p += red[w * 32 + lane];

    // Cross-half reduction on the matrix pipe:
    // A(16x4): A[m,0]=p_m (lanes 0-15), A[m,2]=p_{m+16} (lanes 16-31), cols 1,3 = 0.
    // B(4x16) = all ones  =>  D[m,n] = p_m + p_{m+16} for every n, fp32 exact.
    // Every wave executes the WMMA with EXEC all-ones; only wave 0 stores.
    v2f a; a[0] = p;    a[1] = 0.0f;
    v2f b; b[0] = 1.0f; b[1] = 1.0f;
    v8f c = {};
    v8f dres = __builtin_amdgcn_wmma_f32_16x16x4_f32(
        /*neg_a=*/false, a, /*neg_b=*/false, b,
        /*c_mod=*/(short)0, c, /*reuse_a=*/false, /*reuse_b=*/false);

    const int base = rowTile * ROWTILE;
    if (threadIdx.x == 0) {
#pragma unroll
        for (int v = 0; v < 8; ++v) atomicAdd(&out[base + v], dres[v]);
    } else if (threadIdx.x == 16) {
#pragma unroll
        for (int v = 0; v < 8; ++v) atomicAdd(&out[base + 8 + v], dres[v]);
    }
}

extern "C" void kernel_launch(void* const* d_in, const int* in_sizes, int n_in,
                              void* d_out, int out_size, void* d_ws, size_t ws_size,
                              hipStream_t stream) {
    const float* x      = (const float*)d_in[0];   // [512, 4096]
    const float* unary  = (const float*)d_in[1];   // [4096]
    const float* binary = (const float*)d_in[2];   // [4096, 4096]
    const float* mask   = (const float*)d_in[3];   // [4096, 4096]
    float* out = (float*)d_out;                    // [512]

    float* wr = (float*)d_ws;
    float* wd = wr + NSITES;
    const int use_pre = (ws_size >= (size_t)(2 * NSITES) * sizeof(float)) ? 1 : 0;

    ising_zero<<<1, NBATCH, 0, stream>>>(out);
    if (use_pre)
        ising_precomp<<<(NSITES + 255) / 256, 256, 0, stream>>>(binary, mask, wr, wd);
    ising_energy<<<(NBATCH / ROWTILE) * KCHUNKS, THREADS, 0, stream>>>(
        x, unary, binary, mask, wr, wd, use_pre, out);
}